// FullTensorProductSparse_17798344474677
// MI455X (gfx1250) — compile-verified
//
#include <hip/hip_runtime.h>
#include <hip/hip_bf16.h>

// ---- vector types ----
typedef float v2f  __attribute__((ext_vector_type(2)));
typedef float v8f  __attribute__((ext_vector_type(8)));

#ifndef __has_builtin
#define __has_builtin(x) 0
#endif

#define R2 0.70710678118654752f   // 1/sqrt(2)
#define R3 0.57735026918962576f   // 1/sqrt(3)
#define R6 0.40824829046386302f   // 1/sqrt(6)
#define T6 0.81649658092772603f   // 2/sqrt(6)

// M[k][c] = dot(kT[(k*16+c)], X2), X2 = (s2, vy, vz, vx) (e3nn y,z,x order).
// Channels c: 0 = s*s, 1 = dot/sqrt3, 2-4 = s1*v, 5-7 = w*s2,
//             8-10 = cross/sqrt2, 11-15 = l=2 traceless.
__device__ __constant__ float kT[256] = {
  // k = 0 (pairs with s1): c0..c15, 4 coefs each (s2,vy,vz,vx)
  1,0,0,0,  0,0,0,0,  0,1,0,0,  0,0,1,0,  0,0,0,1,  0,0,0,0,  0,0,0,0,  0,0,0,0,
  0,0,0,0,  0,0,0,0,  0,0,0,0,  0,0,0,0,  0,0,0,0,  0,0,0,0,  0,0,0,0,  0,0,0,0,
  // k = 1 (pairs with w0 = w_y)
  0,0,0,0,  0,R3,0,0, 0,0,0,0,  0,0,0,0,  0,0,0,0,  1,0,0,0,  0,0,0,0,  0,0,0,0,
  0,0,0,0,  0,0,0,-R2, 0,0,R2,0, 0,0,0,R2, 0,0,R2,0, 0,-R6,0,0, 0,0,0,0, 0,-R2,0,0,
  // k = 2 (pairs with w1 = w_z)
  0,0,0,0,  0,0,R3,0, 0,0,0,0,  0,0,0,0,  0,0,0,0,  0,0,0,0,  1,0,0,0,  0,0,0,0,
  0,0,0,R2, 0,0,0,0,  0,-R2,0,0, 0,0,0,0,  0,R2,0,0, 0,0,T6,0, 0,0,0,R2, 0,0,0,0,
  // k = 3 (pairs with w2 = w_x)
  0,0,0,0,  0,0,0,R3, 0,0,0,0,  0,0,0,0,  0,0,0,0,  0,0,0,0,  0,0,0,0,  1,0,0,0,
  0,0,-R2,0, 0,R2,0,0, 0,0,0,0,  0,R2,0,0, 0,0,0,0,  0,0,0,-R6, 0,0,R2,0, 0,0,0,R2
};

// Per-channel output placement: float offset = kBoff[c] + kStep[c]*u.
__device__ __constant__ int kBoff[16] = {
  0, 64, 128, 129, 130, 320, 321, 322, 512, 513, 514, 704, 705, 706, 707, 708
};
__device__ __constant__ int kStep[16] = {
  1, 1, 3, 3, 3, 3, 3, 3, 3, 3, 3, 5, 5, 5, 5, 5
};

// LDS image of one 1024-float output row, padded +4 DWORDs per 128 floats
// (keeps 16B alignment for b128 readback, spreads banks for the scatter).
#define ROWPAD 1056           // 1024 + 8*4
#define WAVES_PER_BLOCK 8

// One wave per sample row n.  Y(64x16) = X1(64x4) @ M(4x16) via four
// V_WMMA_F32_16X16X4_F32.  D layout: lane owns fixed channel c = lane&15,
// 8 u-values u = 16t + 8*(lane>>4) + r  -> uniform 8x ds_store_b32 scatter.
__global__ __launch_bounds__(256) void tp_wmma_kernel(
    const float* __restrict__ in1, const float* __restrict__ in2,
    float* __restrict__ out, int N)
{
  __shared__ float smem[WAVES_PER_BLOCK * ROWPAD];

  const int lane = threadIdx.x & 31;
  const int warp = threadIdx.x >> 5;
  const int wid  = blockIdx.x * WAVES_PER_BLOCK + warp;
  if (wid >= N) return;                 // wave-uniform: EXEC stays all-ones
  const int col  = lane & 15;
  const int half = lane >> 4;

  const float* __restrict__ row1 = in1 + (size_t)wid * 256;
  const float4 x2 = *(const float4*)(in2 + (size_t)wid * 4);
  float* __restrict__ o = out + (size_t)wid * 1024;
  float* __restrict__ ws = smem + warp * ROWPAD;

#if defined(__HIP_DEVICE_COMPILE__) && __has_builtin(__builtin_amdgcn_wmma_f32_16x16x4_f32)
  // ---- build B = M (4x16), branchless: mk[k] = dot4(kT[k][col], X2) ----
  const float4* kT4 = (const float4*)kT;
  float mk[4];
  #pragma unroll
  for (int k = 0; k < 4; ++k) {
    const float4 t = kT4[k * 16 + col];
    mk[k] = t.x * x2.x + t.y * x2.y + t.z * x2.z + t.w * x2.w;
  }
  v2f bm;
  bm.x = half ? mk[2] : mk[0];
  bm.y = half ? mk[3] : mk[1];

  // per-lane scatter parameters (channel c = col)
  const int boff  = kBoff[col];
  const int step  = kStep[col];
  const int step4 = step << 2;                 // byte stride per u

  #pragma unroll
  for (int t = 0; t < 4; ++t) {
    const int ua = t * 16 + col;               // A row this lane supplies
    // A = X1 tile (16x4): lanes 0-15 carry K={0,1} = (s1[ua], w[3ua]),
    // lanes 16-31 carry K={2,3} = (w[3ua+1], w[3ua+2]).
    // 32-bit offset selects -> v_cndmask, no pointer-select divergence.
    const int bw = 64 + 3 * ua;                // w[3ua] position in row1
    const int ox = half ? (bw + 1) : ua;
    const int oy = bw + 2 * half;
    v2f ax; ax.x = row1[ox]; ax.y = row1[oy];

    v8f acc = {0.f, 0.f, 0.f, 0.f, 0.f, 0.f, 0.f, 0.f};
    acc = __builtin_amdgcn_wmma_f32_16x16x4_f32(
        /*neg_a=*/false, ax, /*neg_b=*/false, bm,
        /*c_mod=*/(short)0, acc, /*reuse_a=*/false, /*reuse_b=*/false);

    // uniform scatter: lane's channel c, rows u = 16t + 8*half + r.
    // incremental: idx += step (pad selector), lin += step4 (linear bytes)
    int idx = boff + step * (t * 16 + 8 * half);
    int lin = idx << 2;
    #pragma unroll
    for (int r = 0; r < 8; ++r) {
      *(float*)((char*)ws + lin + ((idx >> 7) << 4)) = acc[r];
      idx += step;
      lin += step4;
    }
  }

  // ---- stream the assembled row to HBM: 8 dense 512B wave stores ----
  #if defined(__HIP_DEVICE_COMPILE__)
  asm volatile("s_wait_dscnt 0x0" ::: "memory");   // scatter visible in LDS
  {
    __attribute__((address_space(3))) float* lbase =
        (__attribute__((address_space(3))) float*)ws;
    unsigned long long ga = (unsigned long long)(o + 4 * lane);
    unsigned int       la = (unsigned int)(size_t)(lbase + 4 * lane);
    #pragma unroll
    for (int i = 0; i < 8; ++i) {
      asm volatile("global_store_async_from_lds_b128 %0, %1, off"
                   :: "v"(ga), "v"(la) : "memory");
      ga += 512;        // 128 floats per chunk in the output row
      la += 528;        // 132 floats per chunk in the padded LDS image
    }
  }
  #endif
#else
  // Scalar fallback (also what the host pass parses): 2 u-values per lane.
  const float s2 = x2.x, vy = x2.y, vz = x2.z, vx = x2.w;
  const float* wvp = row1 + 64;
  for (int q = 0; q < 2; ++q) {
    const int u = lane + 32 * q;
    const float s1 = row1[u];
    const float w0 = wvp[3 * u], w1 = wvp[3 * u + 1], w2 = wvp[3 * u + 2];
    o[u]      = s1 * s2;
    o[64 + u] = R3 * (w0 * vy + w1 * vz + w2 * vx);
    o[128 + 3 * u + 0] = s1 * vy;
    o[128 + 3 * u + 1] = s1 * vz;
    o[128 + 3 * u + 2] = s1 * vx;
    o[320 + 3 * u + 0] = w0 * s2;
    o[320 + 3 * u + 1] = w1 * s2;
    o[320 + 3 * u + 2] = w2 * s2;
    o[512 + 3 * u + 0] = R2 * (w1 * vx - w2 * vz);
    o[512 + 3 * u + 1] = R2 * (w2 * vy - w0 * vx);
    o[512 + 3 * u + 2] = R2 * (w0 * vz - w1 * vy);
    o[704 + 5 * u + 0] = R2 * (w2 * vy + w0 * vx);
    o[704 + 5 * u + 1] = R2 * (w0 * vz + w1 * vy);
    o[704 + 5 * u + 2] = R6 * (2.f * w1 * vz - w2 * vx - w0 * vy);
    o[704 + 5 * u + 3] = R2 * (w1 * vx + w2 * vz);
    o[704 + 5 * u + 4] = R2 * (w2 * vx - w0 * vy);
  }
  (void)ws;
#endif
}

extern "C" void kernel_launch(void* const* d_in, const int* in_sizes, int n_in,
                              void* d_out, int out_size, void* d_ws, size_t ws_size,
                              hipStream_t stream) {
  const float* in1 = (const float*)d_in[0];   // [N, 256] f32
  const float* in2 = (const float*)d_in[1];   // [N, 4]   f32
  float*       out = (float*)d_out;           // [N, 1024] f32
  const int N = in_sizes[0] / 256;
  const int blocks = (N + WAVES_PER_BLOCK - 1) / WAVES_PER_BLOCK;
  tp_wmma_kernel<<<blocks, 256, 0, stream>>>(in1, in2, out, N);
  (void)d_ws; (void)ws_size; (void)n_in; (void)out_size;
}